// earth_mover_distance_49177375539580
// MI455X (gfx1250) — compile-verified
//
#include <hip/hip_runtime.h>
#include <cmath>

// Problem constants (from reference): b=16, n=m=2048, 10 levels.
#define B_    16
#define N_    2048
#define TILES (N_ / 16)     // 128 column tiles of 16
#define TPB   256           // 8 waves (wave32) per workgroup
#define WAVES 8
#define WGS   256           // 16 WGs per batch * 16 batches; each WG owns 128 rows

typedef float v2f __attribute__((ext_vector_type(2)));
typedef float v8f __attribute__((ext_vector_type(8)));

// D = A(16x4,f32) * B(4x16,f32) + C(16x16,f32)  — CDNA5 FP32 WMMA
__device__ __forceinline__ v8f wmma_16x16x4_f32(v2f a, v2f b, v8f c) {
    return __builtin_amdgcn_wmma_f32_16x16x4_f32(
        /*neg_a=*/false, a, /*neg_b=*/false, b,
        /*c_mod=*/(short)0, c, /*reuse_a=*/false, /*reuse_b=*/false);
}

// Pack (x,y,z) -> float4 (x,y,z,|p|^2) so a point is one B128 load and the
// squared norm rides along as the 4th WMMA K-element.
__global__ void pack_kernel(const float* __restrict__ xyz,
                            float4* __restrict__ pk, int total) {
    int t = blockIdx.x * blockDim.x + threadIdx.x;
    if (t < total) {
        float x = xyz[3 * t + 0], y = xyz[3 * t + 1], z = xyz[3 * t + 2];
        pk[t] = make_float4(x, y, z, x * x + y * y + z * z);
    }
}

__global__ void init_kernel(float* __restrict__ remainL,
                            float* __restrict__ remainR,
                            float* __restrict__ costPartial) {
    int t = blockIdx.x * blockDim.x + threadIdx.x;
    if (t < B_ * N_) { remainL[t] = 1.0f; remainR[t] = 1.0f; }  // multiL=multiR=1 (n==m)
    if (t < WGS) costPartial[t] = 0.0f;
}

// One fused sweep over a 2048x2048 exp-distance matrix.
// MODE 0 (pass A): ratioL[i] = remainL[i] / (sum_j K*remainR[j] + 1e-9)
// MODE 1 (pass B): same core, clouds swapped -> col sums; epilogue computes
//                  ratioR[j] and updates remainR[j]
// MODE 2 (pass C): dual row sums p=sum K*ratioR, q=sum K*sqrt(d2)*ratioR;
//                  updates remainL[i] and accumulates cost partials per WG.
// For MODE 0/1 the level scale l2 = level*log2(e) is pre-folded into the A
// fragment and the staged norm2, so the inner loop is add+exp+fma only.
template <int MODE>
__global__ __launch_bounds__(TPB) void pass_kernel(
    const float4* __restrict__ pkRow, const float4* __restrict__ pkCol,
    const float* __restrict__ vec, float* __restrict__ aux1,
    float* __restrict__ aux2, float* __restrict__ costPartial, float l2) {
    __shared__ v2f  sB[2][N_];    // 32 KB: B fragments, pre-split per lane-half
    __shared__ v2f  sNV[N_];      // 16 KB: (scaled |y_j|^2, vec_j)
    __shared__ float sCost[WAVES];

    const int tid   = threadIdx.x;
    const int wg    = blockIdx.x;
    const int batch = wg >> 4;        // 16 WGs per batch
    const int strip = wg & 15;        // 128-row strip within the batch
    const int wave  = tid >> 5;
    const int lane  = tid & 31;
    const int g     = lane & 15;      // column within tile / group lane
    const int half  = lane >> 4;      // A/B fragment half (K split)
    const int rowStart = strip * 128 + wave * 16;
    const int base     = batch * N_;

    // Cooperative LDS stage of this batch's column data (shared by 8 waves).
    const float scaleN = (MODE == 2) ? 1.0f : l2;
    for (int t = tid; t < N_; t += TPB) {
        float4 Q = pkCol[base + t];
        v2f b0; b0.x = -2.0f * Q.x; b0.y = -2.0f * Q.y;   // K=0,1 rows of B
        v2f b1; b1.x = -2.0f * Q.z; b1.y = 1.0f;          // K=2,3 rows of B
        sB[0][t] = b0;
        sB[1][t] = b1;
        v2f nv; nv.x = scaleN * Q.w; nv.y = vec[base + t];
        sNV[t] = nv;
    }
    __syncthreads();

    // A fragment, held for the whole kernel:
    // A[i][0..3] = (x_i, y_i, z_i, |x_i|^2) (times l2 for MODE 0/1).
    // Lanes 0-15 carry K=0,1; lanes 16-31 carry K=2,3.
    float4 P = pkRow[base + rowStart + g];
    v2f afrag;
    afrag.x = half ? P.z : P.x;
    afrag.y = half ? P.w : P.y;
    if (MODE != 2) { afrag.x *= l2; afrag.y *= l2; }

    v8f accP = {0.f, 0.f, 0.f, 0.f, 0.f, 0.f, 0.f, 0.f};
    v8f accQ = {0.f, 0.f, 0.f, 0.f, 0.f, 0.f, 0.f, 0.f};
    const v8f zero = {0.f, 0.f, 0.f, 0.f, 0.f, 0.f, 0.f, 0.f};

#pragma unroll 2
    for (int jt = 0; jt < TILES; ++jt) {
        const int j = jt * 16 + g;
        v2f bfrag = sB[half][j];    // one ds_load_b64, no lane selects
        v2f nv    = sNV[j];         // (scaled norm2, vec)

        v8f c = wmma_16x16x4_f32(afrag, bfrag, zero);

#pragma unroll
        for (int k = 0; k < 8; ++k) {
            if (MODE == 2) {
                float d2 = c[k] + nv.x;                       // raw d2
                float e  = __builtin_amdgcn_exp2f(l2 * d2);   // single v_exp_f32
                accP[k] += e * nv.y;
                float sq = __builtin_amdgcn_sqrtf(fmaxf(d2, 1e-20f));
                accQ[k] += e * sq * nv.y;
            } else {
                // c and nv.x are pre-scaled by l2: exp(level*d2) directly.
                float e = __builtin_amdgcn_exp2f(c[k] + nv.x);
                accP[k] += e * nv.y;
            }
        }
    }

    // Row sums: reduce across the 16 lanes of each half (rows m = k + 8*half).
#pragma unroll
    for (int off = 1; off < 16; off <<= 1) {
#pragma unroll
        for (int k = 0; k < 8; ++k) {
            accP[k] += __shfl_xor(accP[k], off, 16);
            if (MODE == 2) accQ[k] += __shfl_xor(accQ[k], off, 16);
        }
    }

    // Lane g (<8) of each half owns row rowStart + 8*half + g, value acc[g].
    float vP = accP[0], vQ = accQ[0];
#pragma unroll
    for (int kk = 1; kk < 8; ++kk) {
        if (g == kk) { vP = accP[kk]; if (MODE == 2) vQ = accQ[kk]; }
    }

    float cpart = 0.0f;
    if (g < 8) {
        const int r = base + rowStart + 8 * half + g;
        if (MODE == 0) {
            aux2[r] = aux1[r] / (vP + 1e-9f);               // ratioL
        } else if (MODE == 1) {
            float rr   = aux1[r];                           // remainR (old)
            float sumr = vP * rr;
            float cons = fminf(rr / (sumr + 1e-9f), 1.0f);
            aux2[r] = cons * rr;                            // ratioR
            aux1[r] = fmaxf(0.0f, rr - sumr);               // remainR update
        } else {
            float rl = aux1[r];                             // ratioL
            aux2[r]  = fmaxf(0.0f, aux2[r] - rl * vP);      // remainL update
            cpart    = rl * vQ;                             // cost contribution
        }
    }

    if (MODE == 2) {
        // Deterministic per-WG cost partial (no float atomics).
#pragma unroll
        for (int off = 1; off < 32; off <<= 1) cpart += __shfl_xor(cpart, off, 32);
        if (lane == 0) sCost[wave] = cpart;
        __syncthreads();
        if (tid == 0) {
            float s = 0.0f;
            for (int w = 0; w < WAVES; ++w) s += sCost[w];
            costPartial[wg] += s;   // accumulated across the 10 levels
        }
    }
}

__global__ void finalize_kernel(const float* __restrict__ costPartial,
                                float* __restrict__ out) {
    __shared__ float s[WGS];
    int t = threadIdx.x;
    s[t] = costPartial[t];
    __syncthreads();
    for (int off = WGS / 2; off > 0; off >>= 1) {
        if (t < off) s[t] += s[t + off];
        __syncthreads();
    }
    if (t == 0) out[0] = s[0] / ((float)N_ * (float)B_);  // mean(cost / n)
}

extern "C" void kernel_launch(void* const* d_in, const int* in_sizes, int n_in,
                              void* d_out, int out_size, void* d_ws, size_t ws_size,
                              hipStream_t stream) {
    const float* xyz1 = (const float*)d_in[0];
    const float* xyz2 = (const float*)d_in[1];
    float* out = (float*)d_out;

    // Workspace layout (~1.5 MB total)
    float4* pk1 = (float4*)d_ws;              // B*N float4
    float4* pk2 = pk1 + B_ * N_;              // B*N float4
    float* remainL     = (float*)(pk2 + B_ * N_);
    float* remainR     = remainL + B_ * N_;
    float* ratioL      = remainR + B_ * N_;
    float* ratioR      = ratioL  + B_ * N_;
    float* costPartial = ratioR  + B_ * N_;   // WGS floats

    const int blocks = (B_ * N_ + TPB - 1) / TPB;
    pack_kernel<<<blocks, TPB, 0, stream>>>(xyz1, pk1, B_ * N_);
    pack_kernel<<<blocks, TPB, 0, stream>>>(xyz2, pk2, B_ * N_);
    init_kernel<<<blocks, TPB, 0, stream>>>(remainL, remainR, costPartial);

    const double LOG2E = 1.4426950408889634;
    for (int j = 0; j < 10; ++j) {
        float level = (j < 9) ? (float)(-pow(4.0, 7 - j)) : 0.0f;
        float l2 = (float)((double)level * LOG2E);
        // Pass A: a = K @ remainR -> ratioL   (rows = cloud1, cols = cloud2)
        pass_kernel<0><<<WGS, TPB, 0, stream>>>(pk1, pk2, remainR, remainL,
                                                ratioL, nullptr, l2);
        // Pass B: s = K^T @ ratioL -> ratioR, remainR  (clouds swapped)
        pass_kernel<1><<<WGS, TPB, 0, stream>>>(pk2, pk1, ratioL, remainR,
                                                ratioR, nullptr, l2);
        // Pass C: row sums of K*ratioR and K*sqrt(d2)*ratioR -> remainL, cost
        pass_kernel<2><<<WGS, TPB, 0, stream>>>(pk1, pk2, ratioR, ratioL,
                                                remainL, costPartial, l2);
    }
    finalize_kernel<<<1, WGS, 0, stream>>>(costPartial, out);
}